// MatrixRouting_40089224741388
// MI455X (gfx1250) — compile-verified
//
#include <hip/hip_runtime.h>
#include <math.h>

typedef __attribute__((ext_vector_type(2))) float v2f;
typedef __attribute__((ext_vector_type(4))) float v4f;
typedef __attribute__((ext_vector_type(8))) float v8f;

#define BATCH 8
#define BKK   288
#define CWW   1152
#define HH    17
#define HD    16
#define CDIM  32
#define WW2   36
#define EPSF  1e-10f
#define HALF_LN_2PI 0.9189385332046727f

__device__ __forceinline__ size_t vidx(int b, int i, int j, int h) {
    return ((size_t)((b * BKK + i) * CWW + j)) * HH + h;
}

__global__ void k_init_R(float* __restrict__ R, int n) {
    int t = blockIdx.x * blockDim.x + threadIdx.x;
    if (t < n) R[t] = 1.0f / 32.0f;
}

// Pass 1: Rp = R * a_, sum_R = sum_i Rp + 1e-4, mu = (sum_i Rp*V)/sum_R.
// One wave per (b, j). The i-reduction runs on the matrix pipe:
// A = ones(16x4), B[k, n=h] = Rp[i]*V[i, j, h]; with A==1 the result
// D[m,n] = sum_k B[k,n] is invariant to K-row layout permutations.
// i unrolled by 8 -> 2 WMMAs per iteration; addresses are base+const offsets.
__global__ void k_mu(const float* __restrict__ votes, float* __restrict__ R,
                     float* __restrict__ mu, float* __restrict__ sumR)
{
    const int lane = threadIdx.x & 31;
    const int warp = threadIdx.x >> 5;
    const int pair = blockIdx.x * 8 + warp;          // 0..9215 == (b, j)
    const int b   = pair / CWW;
    const int j   = pair % CWW;
    const int col = lane & 15;                        // h column
    const int rsel = lane >> 4;                       // which K row this half-wave feeds

    const int RS = CWW * HH;                          // votes i-stride = 19584 elems
    const float* pV = votes + (size_t)(b * BKK + rsel) * RS + (size_t)j * HH;
    float*       pR = R     + (size_t)(b * BKK + rsel) * CWW + j;

    const v2f aones = {1.0f, 1.0f};
    v8f acc = {};
    float racc = 0.0f;

    for (int i0 = 0; i0 < BKK; i0 += 8) {
        __builtin_prefetch(pV + 8 * RS, 0, 3);        // global_prefetch_b8 (cold first sweep)
        const float r0 = pR[0 * CWW] * pV[0 * RS + HD];
        const float r1 = pR[2 * CWW] * pV[2 * RS + HD];
        const float r2 = pR[4 * CWW] * pV[4 * RS + HD];
        const float r3 = pR[6 * CWW] * pV[6 * RS + HD];
        v2f b0, b1;
        b0[0] = r0 * pV[0 * RS + col];
        b0[1] = r1 * pV[2 * RS + col];
        b1[0] = r2 * pV[4 * RS + col];
        b1[1] = r3 * pV[6 * RS + col];
        // Persist Rp (R <- R*a_), in place. All 16 lanes of each half-wave
        // write the same value to the same address: deterministic, branch-free.
        pR[0 * CWW] = r0;
        pR[2 * CWW] = r1;
        pR[4 * CWW] = r2;
        pR[6 * CWW] = r3;
        racc += (r0 + r1) + (r2 + r3);                // identical within each half-wave
        acc = __builtin_amdgcn_wmma_f32_16x16x4_f32(false, aones, false, b0,
                                                    (short)0, acc, false, false);
        acc = __builtin_amdgcn_wmma_f32_16x16x4_f32(false, aones, false, b1,
                                                    (short)0, acc, false, false);
        pV += 8 * RS;
        pR += 8 * CWW;
    }
    const float s = __shfl(racc, 0, 32) + __shfl(racc, 16, 32) + 1e-4f;
    if (lane == 0) sumR[b * CWW + j] = s;
    if (lane < 16) mu[(size_t)(b * CWW + j) * HD + lane] = acc[0] / s;
}

// Pass 2: sigma_square = (sum_i Rp*((V-mu)^2+eps))/sum_R + eps, same WMMA reduction.
__global__ void k_sigma(const float* __restrict__ votes, const float* __restrict__ R,
                        const float* __restrict__ mu, const float* __restrict__ sumR,
                        float* __restrict__ sig2)
{
    const int lane = threadIdx.x & 31;
    const int warp = threadIdx.x >> 5;
    const int pair = blockIdx.x * 8 + warp;
    const int b   = pair / CWW;
    const int j   = pair % CWW;
    const int col = lane & 15;
    const int rsel = lane >> 4;

    const int RS = CWW * HH;
    const float* pV = votes + (size_t)(b * BKK + rsel) * RS + (size_t)j * HH;
    const float* pR = R     + (size_t)(b * BKK + rsel) * CWW + j;

    const float muv = mu[(size_t)(b * CWW + j) * HD + col];
    const v2f aones = {1.0f, 1.0f};
    v8f acc = {};

    for (int i0 = 0; i0 < BKK; i0 += 8) {
        __builtin_prefetch(pV + 8 * RS, 0, 3);
        const float r0 = pR[0 * CWW];
        const float r1 = pR[2 * CWW];
        const float r2 = pR[4 * CWW];
        const float r3 = pR[6 * CWW];
        const float d0 = pV[0 * RS + col] - muv;
        const float d1 = pV[2 * RS + col] - muv;
        const float d2 = pV[4 * RS + col] - muv;
        const float d3 = pV[6 * RS + col] - muv;
        v2f b0, b1;
        b0[0] = r0 * (d0 * d0 + EPSF);
        b0[1] = r1 * (d1 * d1 + EPSF);
        b1[0] = r2 * (d2 * d2 + EPSF);
        b1[1] = r3 * (d3 * d3 + EPSF);
        acc = __builtin_amdgcn_wmma_f32_16x16x4_f32(false, aones, false, b0,
                                                    (short)0, acc, false, false);
        acc = __builtin_amdgcn_wmma_f32_16x16x4_f32(false, aones, false, b1,
                                                    (short)0, acc, false, false);
        pV += 8 * RS;
        pR += 8 * CWW;
    }
    if (lane < 16) {
        const float s = sumR[b * CWW + j];
        sig2[(size_t)(b * CWW + j) * HD + lane] = acc[0] / s + EPSF;
    }
}

// Activation: a = sigmoid(lambda * (beta_a - sum_h (beta_v + log_sigma) * sum_R))
__global__ void k_act(const float* __restrict__ sig2, const float* __restrict__ sumR,
                      const float* __restrict__ beta_v, const float* __restrict__ beta_a,
                      float* __restrict__ aout)
{
    int t = blockIdx.x * blockDim.x + threadIdx.x;
    if (t >= BATCH * CWW) return;
    const int j = t % CWW;
    const int c = j / WW2;
    const float bv = beta_v[c];
    float cost = 0.0f;
    for (int h = 0; h < HD; ++h) {
        const float s2 = sig2[(size_t)t * HD + h];
        const float ls = logf(sqrtf(s2) + EPSF);
        cost += bv + ls;
    }
    cost *= sumR[t];
    const float x = 1e-4f * (beta_a[c] - cost);
    aout[t] = 1.0f / (1.0f + expf(-x));
}

// E-step (iters 0,1): one block per (b,i); ap[j] = a_j * sum_h exp(ln p),
// normalize over j, write R.
__global__ void k_update(const float* __restrict__ votes, const float* __restrict__ mu,
                         const float* __restrict__ sig2, const float* __restrict__ aout,
                         float* __restrict__ R)
{
    __shared__ float ap[CWW];
    __shared__ float red[8];
    const int bi = blockIdx.x;                 // 0..2303
    const int b = bi / BKK;
    const int i = bi % BKK;
    const int tid = threadIdx.x;

    for (int j = tid; j < CWW; j += 256) {
        float s = 0.0f;
        const size_t mbase = (size_t)(b * CWW + j) * HD;
        const size_t vbase = vidx(b, i, j, 0);
        for (int h = 0; h < HD; ++h) {
            const float v  = votes[vbase + h];
            const float m  = mu[mbase + h];
            const float s2 = sig2[mbase + h];
            const float ls = logf(sqrtf(s2) + EPSF);
            const float d  = v - m;
            const float vms = d * d + EPSF;
            s += expf(-vms / (2.0f * s2) - ls - HALF_LN_2PI);
        }
        ap[j] = aout[b * CWW + j] * s;
    }
    __syncthreads();
    float part = 0.0f;
    for (int j = tid; j < CWW; j += 256) part += ap[j];
    for (int off = 16; off > 0; off >>= 1) part += __shfl_down(part, off, 32);
    if ((tid & 31) == 0) red[tid >> 5] = part;
    __syncthreads();
    if (tid == 0) {
        float tot = 0.0f;
        for (int w = 0; w < 8; ++w) tot += red[w];
        red[0] = tot;
    }
    __syncthreads();
    const float tot = red[0] + EPSF;
    for (int j = tid; j < CWW; j += 256) {
        R[(size_t)(b * BKK + i) * CWW + j] = ap[j] / tot + EPSF;
    }
}

// Assemble mu_a: (b, C, 36, 17) with h<16 -> mu, h==16 -> a.
__global__ void k_mua(const float* __restrict__ mu, const float* __restrict__ aout,
                      float* __restrict__ out)
{
    int t = blockIdx.x * blockDim.x + threadIdx.x;
    if (t >= BATCH * CDIM * WW2 * HH) return;
    const int h  = t % HH;
    const int q  = t / HH;
    const int ww = q % WW2;
    const int q2 = q / WW2;
    const int c  = q2 % CDIM;
    const int b  = q2 / CDIM;
    const int j  = c * WW2 + ww;
    out[t] = (h < HD) ? mu[(size_t)(b * CWW + j) * HD + h] : aout[b * CWW + j];
}

// Non-temporal bulk copy: these outputs are write-once / never re-read, and the
// source (votes) has had its last compute use -> don't pollute the 192MB L2.
// Uses native clang vectors (ext_vector_type) for the nontemporal builtins.
__global__ void k_copy4(const v4f* __restrict__ src, v4f* __restrict__ dst, int n4) {
    int t = blockIdx.x * blockDim.x + threadIdx.x;
    if (t < n4) {
        v4f v = __builtin_nontemporal_load(src + t);
        __builtin_nontemporal_store(v, dst + t);
    }
}

extern "C" void kernel_launch(void* const* d_in, const int* in_sizes, int n_in,
                              void* d_out, int out_size, void* d_ws, size_t ws_size,
                              hipStream_t stream) {
    (void)in_sizes; (void)n_in; (void)out_size; (void)ws_size;
    const float* votes  = (const float*)d_in[0];
    const float* beta_v = (const float*)d_in[1];
    const float* beta_a = (const float*)d_in[2];
    float* out = (float*)d_out;

    float* ws   = (float*)d_ws;
    float* R    = ws;                                   // 8*288*1152
    float* mu   = R    + (size_t)BATCH * BKK * CWW;     // 8*1152*16
    float* sig2 = mu   + (size_t)BATCH * CWW * HD;      // 8*1152*16
    float* sumR = sig2 + (size_t)BATCH * CWW * HD;      // 8*1152
    float* aout = sumR + (size_t)BATCH * CWW;           // 8*1152

    const int nR = BATCH * BKK * CWW;                   // 2,654,208
    k_init_R<<<(nR + 255) / 256, 256, 0, stream>>>(R, nR);

    for (int it = 0; it < 3; ++it) {
        k_mu   <<<BATCH * CWW / 8, 256, 0, stream>>>(votes, R, mu, sumR);
        k_sigma<<<BATCH * CWW / 8, 256, 0, stream>>>(votes, R, mu, sumR, sig2);
        k_act  <<<(BATCH * CWW + 255) / 256, 256, 0, stream>>>(sig2, sumR, beta_v, beta_a, aout);
        if (it < 2)
            k_update<<<BATCH * BKK, 256, 0, stream>>>(votes, mu, sig2, aout, R);
    }

    const int n_mua = BATCH * CDIM * WW2 * HH;          // 156,672
    k_mua<<<(n_mua + 255) / 256, 256, 0, stream>>>(mu, aout, out);

    float* outR = out + n_mua;                          // R_out = Rp of iteration 2
    k_copy4<<<(nR / 4 + 255) / 256, 256, 0, stream>>>((const v4f*)R, (v4f*)outR, nR / 4);

    float* outV = outR + nR;                            // votes passthrough
    const int nV4 = BATCH * BKK * CWW * HH / 4;         // 11,280,384
    k_copy4<<<(nV4 + 255) / 256, 256, 0, stream>>>((const v4f*)votes, (v4f*)outV, nV4);
}